// Net_66606352826792
// MI455X (gfx1250) — compile-verified
//
#include <hip/hip_runtime.h>
#include <hip/hip_bf16.h>

typedef __attribute__((ext_vector_type(2))) float v2f;
typedef __attribute__((ext_vector_type(8))) float v8f;

// ---------------------------------------------------------------------------
// Dual GEMM with fp32 WMMA: A = X @ W[:CIN] + bias, V = X @ W[CIN:]
// X: (rows, CIN) row-major, W: (2*CIN, COUT) row-major.
// One wave (32 threads) per 16x16 output tile; K consumed 4 at a time by
// V_WMMA_F32_16X16X4_F32.
// ---------------------------------------------------------------------------
template<int CIN, int COUT>
__global__ __launch_bounds__(32) void edge_gemm_dual(
    const float* __restrict__ X,
    const float* __restrict__ W,
    const float* __restrict__ bias,
    float* __restrict__ A,
    float* __restrict__ V)
{
    constexpr int NT = (COUT + 15) / 16;     // tiles along output channels
    const int lane = threadIdx.x;            // 0..31, wave32
    const int mt = blockIdx.x / NT;
    const int nt = blockIdx.x % NT;
    const int m0 = mt * 16;
    const int n0 = nt * 16;
    const int half = lane >> 4;              // 0: K lo pair, 1: K hi pair
    const int l15  = lane & 15;

    const int rowA = m0 + l15;               // A-frag: lane&15 = M
    const int colB = n0 + l15;               // B-frag: lane&15 = N
    const bool cok = (colB < COUT);

    v8f accA = {};
    v8f accV = {};

    for (int kk = 0; kk < CIN; kk += 4) {
        const int ka = kk + half * 2;        // this lane's K base
        v2f a;
        a.x = (ka     < CIN) ? X[rowA * CIN + ka]     : 0.0f;
        a.y = (ka + 1 < CIN) ? X[rowA * CIN + ka + 1] : 0.0f;

        v2f bt, bb;
        bt.x = (cok && ka     < CIN) ? W[ ka            * COUT + colB] : 0.0f;
        bt.y = (cok && ka + 1 < CIN) ? W[(ka + 1)       * COUT + colB] : 0.0f;
        bb.x = (cok && ka     < CIN) ? W[(CIN + ka)     * COUT + colB] : 0.0f;
        bb.y = (cok && ka + 1 < CIN) ? W[(CIN + ka + 1) * COUT + colB] : 0.0f;

        // (neg_a, A, neg_b, B, c_mod, C, reuse_a, reuse_b)
        accA = __builtin_amdgcn_wmma_f32_16x16x4_f32(
                   false, a, false, bt, (short)0, accA, false, false);
        accV = __builtin_amdgcn_wmma_f32_16x16x4_f32(
                   false, a, false, bb, (short)0, accV, false, false);
    }

    if (cok) {
        const float bv = bias[colB];
#pragma unroll
        for (int r = 0; r < 8; ++r) {
            const int row = m0 + r + half * 8;   // D layout: VGPR r -> M=r / M=8+r
            A[row * COUT + colB] = accA[r] + bv;
            V[row * COUT + colB] = accV[r];
        }
    }
}

// ---------------------------------------------------------------------------
// Pairwise reduction over all j in the same batch:
//   out[i,c] = ( sum_j relu(A[i,c] + V[j,c]) - relu(A[i,c] + V[i,c]) ) / (N-1)
// One thread per (i, c); V staged through LDS in tiles of TJ rows.
// ---------------------------------------------------------------------------
template<int C>
__global__ __launch_bounds__(256) void edge_pairwise_relu(
    const float* __restrict__ A,
    const float* __restrict__ V,
    float* __restrict__ out,
    int N, float inv_nm1)
{
    constexpr int TJ = 128;                  // j-tile rows
    constexpr int R  = 256 / C;              // i rows per block
    __shared__ float vt[TJ * C];             // <= 32 KB

    const int tid  = threadIdx.x;
    const int c    = tid % C;
    const int iloc = tid / C;
    const int i    = blockIdx.x * R + iloc;
    const int base = (i / N) * N;            // batch start row (R divides N)

    const float a = A[i * C + c];
    float selfterm = fmaxf(a + V[i * C + c], 0.0f);

    float s = 0.0f;
    for (int jt = 0; jt < N; jt += TJ) {
        const float* src = V + (size_t)(base + jt) * C;
        for (int t = tid; t < TJ * C; t += 256) vt[t] = src[t];
        __syncthreads();
#pragma unroll 8
        for (int jj = 0; jj < TJ; ++jj)
            s += fmaxf(a + vt[jj * C + c], 0.0f);
        __syncthreads();
    }
    out[i * C + c] = (s - selfterm) * inv_nm1;
}

// ---------------------------------------------------------------------------
// Final (no-ReLU) block closed form: out = A + (colsum(V) - V) / (N-1)
// ---------------------------------------------------------------------------
__global__ __launch_bounds__(256) void edge_colsum3(
    const float* __restrict__ V, float* __restrict__ cs, int N)
{
    __shared__ float red[256];
    const int b = blockIdx.x / 3;
    const int c = blockIdx.x % 3;
    float s = 0.0f;
    for (int j = threadIdx.x; j < N; j += 256)
        s += V[(size_t)(b * N + j) * 3 + c];
    red[threadIdx.x] = s;
    __syncthreads();
    for (int off = 128; off > 0; off >>= 1) {
        if (threadIdx.x < off) red[threadIdx.x] += red[threadIdx.x + off];
        __syncthreads();
    }
    if (threadIdx.x == 0) cs[blockIdx.x] = red[0];
}

__global__ __launch_bounds__(256) void edge_final3(
    const float* __restrict__ A,
    const float* __restrict__ V,
    const float* __restrict__ cs,
    float* __restrict__ out,
    int N, float inv_nm1)
{
    const int idx = blockIdx.x * 256 + threadIdx.x;    // < 2*N*3
    const int c = idx % 3;
    const int b = (idx / 3) / N;
    out[idx] = A[idx] + (cs[b * 3 + c] - V[idx]) * inv_nm1;
}

// ---------------------------------------------------------------------------
// Host-side orchestration
// ---------------------------------------------------------------------------
extern "C" void kernel_launch(void* const* d_in, const int* in_sizes, int n_in,
                              void* d_out, int out_size, void* d_ws, size_t ws_size,
                              hipStream_t stream) {
    (void)in_sizes; (void)n_in; (void)out_size; (void)ws_size;

    const float* x  = (const float*)d_in[0];   // (2, 3072) == (2048, 3) row-major
    const float* W1 = (const float*)d_in[1];   // (6, 32)
    const float* b1 = (const float*)d_in[2];   // (32,)
    const float* W2 = (const float*)d_in[3];   // (64, 64)
    const float* b2 = (const float*)d_in[4];   // (64,)
    const float* W3 = (const float*)d_in[5];   // (128, 3)
    const float* b3 = (const float*)d_in[6];   // (3,)
    float* out = (float*)d_out;

    const int B = 2, N = 1024;
    const int ROWS = B * N;                    // 2048
    const float inv = 1.0f / (float)(N - 1);

    // Workspace layout (floats); total ~2.35 MB
    float* ws = (float*)d_ws;
    float* A1 = ws;                 // 2048*32
    float* V1 = A1 + ROWS * 32;     // 2048*32
    float* H1 = V1 + ROWS * 32;     // 2048*32
    float* A2 = H1 + ROWS * 32;     // 2048*64
    float* V2 = A2 + ROWS * 64;     // 2048*64
    float* H2 = V2 + ROWS * 64;     // 2048*64
    float* A3 = H2 + ROWS * 64;     // 2048*3
    float* V3 = A3 + ROWS * 3;      // 2048*3
    float* CS = V3 + ROWS * 3;      // 6

    const int MT = ROWS / 16;                  // 128 row tiles

    // Block 1: 3 -> 32, ReLU
    edge_gemm_dual<3, 32><<<MT * 2, 32, 0, stream>>>(x, W1, b1, A1, V1);
    edge_pairwise_relu<32><<<ROWS / (256 / 32), 256, 0, stream>>>(A1, V1, H1, N, inv);

    // Block 2: 32 -> 64, ReLU
    edge_gemm_dual<32, 64><<<MT * 4, 32, 0, stream>>>(H1, W2, b2, A2, V2);
    edge_pairwise_relu<64><<<ROWS / (256 / 64), 256, 0, stream>>>(A2, V2, H2, N, inv);

    // Block 3: 64 -> 3, linear => closed form
    edge_gemm_dual<64, 3><<<MT * 1, 32, 0, stream>>>(H2, W3, b3, A3, V3);
    edge_colsum3<<<B * 3, 256, 0, stream>>>(V3, CS, N);
    edge_final3<<<(ROWS * 3) / 256, 256, 0, stream>>>(A3, V3, CS, out, N, inv);
}